// DynamicVoxelEncoderV1_60009283060241
// MI455X (gfx1250) — compile-verified
//
#include <hip/hip_runtime.h>
#include <hip/hip_bf16.h>
#include <stdint.h>

// DynamicVoxelEncoder scatter-mean for MI455X (gfx1250).
// Strategy: dense 64^3 accumulator (8 MB, lives in L2) + f32 atomics,
// then prefix-scan compaction to emit sorted-unique voxel means.
// CDNA5-specific paths: global_load_async_to_lds_b128 staging (ASYNCcnt),
// s_wait_asynccnt, wave32 ballot scans, nontemporal wide fills.

#define GDIM 64
#define NVOX (GDIM * GDIM * GDIM)   // 262144 voxels
#define TPB  256
#define NBLK (NVOX / TPB)           // 1024

typedef uint32_t v4u __attribute__((ext_vector_type(4)));

__device__ __forceinline__ uint32_t ballot32(int p) {
#if __has_builtin(__builtin_amdgcn_ballot_w32)
  return __builtin_amdgcn_ballot_w32(p != 0);
#else
  return (uint32_t)__ballot(p);
#endif
}

// ---------------------------------------------------------------------------
// Fill kernel: nontemporal 128-bit stores so the 160 MB of pad-fill does not
// evict the 8 MB accumulator from the 192 MB L2.
// ---------------------------------------------------------------------------
__global__ void __launch_bounds__(TPB)
fill_u32_kernel(uint32_t* __restrict__ p, size_t n, uint32_t val) {
  size_t i = (size_t)blockIdx.x * blockDim.x + threadIdx.x;
  size_t stride = (size_t)gridDim.x * blockDim.x;
  size_t n4 = n >> 2;
  v4u v = {val, val, val, val};
  for (size_t k = i; k < n4; k += stride)
    __builtin_nontemporal_store(v, ((v4u*)p) + k);
  for (size_t k = (n4 << 2) + i; k < n; k += stride)
    p[k] = val;
}

// ---------------------------------------------------------------------------
// Scatter kernel: stage the 28-byte point rows into LDS with the CDNA5 async
// DMA path (perfectly coalesced b128 streams, tracked by ASYNCcnt), then do
// 8 f32 atomics per point into the dense accumulator (resolved in L2).
// LDS row stride is 7 dwords (odd) -> conflict-free across 64 banks.
// ---------------------------------------------------------------------------
__global__ void __launch_bounds__(TPB)
scatter_kernel(const float* __restrict__ points, const int* __restrict__ grid,
               float* __restrict__ acc, int N) {
  __shared__ float tile[TPB * 7];
  const int tid = threadIdx.x;
  const long long base = (long long)blockIdx.x * TPB;
  long long remRows = (long long)N - base;
  const int rows = (remRows < TPB) ? (int)remRows : TPB;
  const int nfl = rows * 7;                       // floats to stage
  const float* src = points + base * 7;
  const uint32_t lds0 = (uint32_t)(uintptr_t)(&tile[0]);  // LDS byte offset

  // Async copy: full 16-byte chunks.
  const int c4 = nfl >> 2;
  for (int c = tid; c < c4; c += TPB) {
    uint32_t loff = lds0 + ((uint32_t)c << 4);
    uint32_t goff = (uint32_t)c << 4;
    asm volatile("global_load_async_to_lds_b128 %0, %1, %2"
                 :: "v"(loff), "v"(goff), "s"(src) : "memory");
  }
  // Tail floats (0..3) via b32 async copies.
  for (int r = (c4 << 2) + tid; r < nfl; r += TPB) {
    uint32_t loff = lds0 + ((uint32_t)r << 2);
    uint32_t goff = (uint32_t)r << 2;
    asm volatile("global_load_async_to_lds_b32 %0, %1, %2"
                 :: "v"(loff), "v"(goff), "s"(src) : "memory");
  }
  asm volatile("s_wait_asynccnt 0x0" ::: "memory");
  __syncthreads();

  const long long i = base + tid;
  if (i < (long long)N) {
    float p[7];
#pragma unroll
    for (int c = 0; c < 7; ++c) p[c] = tile[tid * 7 + c];
    const int* g = grid + i * 3;
    const int lin = (g[0] * GDIM + g[1]) * GDIM + g[2];
    float* a = acc + (size_t)lin * 8;
#pragma unroll
    for (int c = 0; c < 7; ++c) atomicAdd(a + c, p[c]);
    atomicAdd(a + 7, 1.0f);
  }
}

// ---------------------------------------------------------------------------
// Stage 1 of occupancy scan: per-256-voxel block totals (wave32 ballot).
// ---------------------------------------------------------------------------
__global__ void __launch_bounds__(TPB)
block_count_kernel(const float* __restrict__ acc, uint32_t* __restrict__ totals) {
  const int v = blockIdx.x * TPB + threadIdx.x;
  const int occ = acc[(size_t)v * 8 + 7] > 0.0f;
  const uint32_t m = ballot32(occ);
  const int lane = threadIdx.x & 31, wv = threadIdx.x >> 5;
  __shared__ uint32_t wsum[TPB / 32];
  if (lane == 0) wsum[wv] = (uint32_t)__popc(m);
  __syncthreads();
  if (threadIdx.x == 0) {
    uint32_t s = 0;
#pragma unroll
    for (int k = 0; k < TPB / 32; ++k) s += wsum[k];
    totals[blockIdx.x] = s;
  }
}

// ---------------------------------------------------------------------------
// Stage 2: exclusive scan of the 1024 block totals (single workgroup).
// ---------------------------------------------------------------------------
__global__ void __launch_bounds__(NBLK)
scan_kernel(const uint32_t* __restrict__ in, uint32_t* __restrict__ out) {
  __shared__ uint32_t s[NBLK];
  const int t = threadIdx.x;
  const uint32_t x = in[t];
  s[t] = x;
  __syncthreads();
  for (int off = 1; off < NBLK; off <<= 1) {
    uint32_t v = (t >= off) ? s[t - off] : 0u;
    __syncthreads();
    s[t] += v;
    __syncthreads();
  }
  out[t] = s[t] - x;  // exclusive prefix
}

// ---------------------------------------------------------------------------
// Stage 3: emit sorted-unique voxel means + coords at their global rank.
// Ascending voxel id == torch.unique sorted order.
// ---------------------------------------------------------------------------
__global__ void __launch_bounds__(TPB)
emit_kernel(const float* __restrict__ acc, const uint32_t* __restrict__ offs,
            float* __restrict__ outMean, float* __restrict__ outCoord) {
  const int v = blockIdx.x * TPB + threadIdx.x;
  const float cnt = acc[(size_t)v * 8 + 7];
  const int occ = cnt > 0.0f;
  const uint32_t m = ballot32(occ);
  const int lane = threadIdx.x & 31, wv = threadIdx.x >> 5;
  __shared__ uint32_t wsum[TPB / 32];
  if (lane == 0) wsum[wv] = (uint32_t)__popc(m);
  __syncthreads();
  uint32_t wpre = 0;
  for (int k = 0; k < wv; ++k) wpre += wsum[k];
  if (occ) {
    const uint32_t rank =
        offs[blockIdx.x] + wpre + (uint32_t)__popc(m & ((1u << lane) - 1u));
    const float* a = acc + (size_t)v * 8;
    float* o = outMean + (size_t)rank * 7;
#pragma unroll
    for (int c = 0; c < 7; ++c) o[c] = a[c] / cnt;
    float* q = outCoord + (size_t)rank * 3;
    q[0] = (float)(v >> 12);
    q[1] = (float)((v >> 6) & 63);
    q[2] = (float)(v & 63);
  }
}

// ---------------------------------------------------------------------------
extern "C" void kernel_launch(void* const* d_in, const int* in_sizes, int n_in,
                              void* d_out, int out_size, void* d_ws, size_t ws_size,
                              hipStream_t stream) {
  const float* points = (const float*)d_in[0];       // [N,7] f32
  const int*   grid   = (const int*)d_in[1];         // [N,3] int (per harness contract)
  const int N = in_sizes[0] / 7;

  float* outMean  = (float*)d_out;                   // [N,7] f32
  float* outCoord = outMean + (size_t)N * 7;         // [N,3] coords (f32 slots)

  // Workspace: dense accumulator 262144*8 f32 (8 MB) + scan scratch (8 KB).
  float*    acc    = (float*)d_ws;
  uint32_t* totals = (uint32_t*)((char*)d_ws + (size_t)NVOX * 8 * sizeof(float));
  uint32_t* offs   = totals + NBLK;

  // 1) zero the accumulator (stays resident in L2)
  fill_u32_kernel<<<1024, TPB, 0, stream>>>((uint32_t*)acc, (size_t)NVOX * 8, 0u);
  // 2) pad-fill outputs: mean rows = 0.0f, coord rows = -1.0f (NT stores)
  fill_u32_kernel<<<4096, TPB, 0, stream>>>((uint32_t*)outMean, (size_t)N * 7, 0u);
  fill_u32_kernel<<<4096, TPB, 0, stream>>>((uint32_t*)outCoord, (size_t)N * 3,
                                            0xBF800000u /* -1.0f */);
  // 3) scatter-accumulate (async-LDS staged point reads, L2 f32 atomics)
  scatter_kernel<<<(N + TPB - 1) / TPB, TPB, 0, stream>>>(points, grid, acc, N);
  // 4) occupancy ranks (two-level scan over the dense voxel array)
  block_count_kernel<<<NBLK, TPB, 0, stream>>>(acc, totals);
  scan_kernel<<<1, NBLK, 0, stream>>>(totals, offs);
  // 5) compact: write means + decoded coords at sorted-unique rank
  emit_kernel<<<NBLK, TPB, 0, stream>>>(acc, offs, outMean, outCoord);
}